// FFMLayer_26749056320145
// MI455X (gfx1250) — compile-verified
//
#include <hip/hip_runtime.h>
#include <hip/hip_bf16.h>

// ---- problem constants (match reference) ----
static constexpr int V_DEPTH  = 10000;
static constexpr int N_DENSE  = 13;
static constexpr int N_SPARSE = 26;
static constexpr int FIELD    = 39;            // 13 + 26
static constexpr int K_DIM    = 8;
static constexpr int TK       = FIELD * K_DIM; // 312 floats per v-row
static constexpr int BATCH    = 4096;

static constexpr int ROWS_PER_BLOCK = 16;      // one WMMA M-tile
static constexpr int THREADS        = 640;     // 20 wave32 -> 20 N-tiles of 16
static constexpr int NT_PAD         = 320;     // padded N (312 -> 320)
static constexpr int BSTRIDE        = 17;      // LDS B stride (bank-conflict free)

typedef __attribute__((ext_vector_type(2))) float v2f;
typedef __attribute__((ext_vector_type(8))) float v8f;

__global__ __launch_bounds__(THREADS)
void ffm_kernel(const float* __restrict__ dense,   // [B,13]
                const int*   __restrict__ sparse,  // [B,26]
                const float* __restrict__ w0,      // [1]
                const float* __restrict__ w,       // [FEAT]
                const float* __restrict__ v,       // [FEAT, 39, 8] row-major
                float*       __restrict__ out)     // [B]
{
    __shared__ float s_densA[ROWS_PER_BLOCK * 16];       // 16x16 A tile (K padded)
    __shared__ int   s_idx[ROWS_PER_BLOCK * N_SPARSE];   // flat feature ids
    __shared__ float s_lin[ROWS_PER_BLOCK];              // linear part per row
    __shared__ float s_dp[ROWS_PER_BLOCK * NT_PAD];      // WMMA dense_part [16][320]
    // s_B (phase 1) overlaid with s_red/s_part (phase 2); phases separated by barriers.
    __shared__ float s_scratch[NT_PAD * BSTRIDE];        // 5440 floats

    float* const s_B    = s_scratch;                     // [tcol*17 + d], 320x16 used
    float* const s_red  = s_scratch;                     // 2 pairs x (312 ff + 312 ff^2)
    float* const s_part = s_scratch + 4096;              // 2 pairs x 48

    const int t    = threadIdx.x;
    const int row0 = blockIdx.x * ROWS_PER_BLOCK;

    // ---------------- Phase 0a: stage A tile, padded B, gather indices -------
    if (t < ROWS_PER_BLOCK * 16) {
        const int m = t >> 4, d = t & 15;
        s_densA[t] = (d < N_DENSE) ? dense[(size_t)(row0 + m) * N_DENSE + d] : 0.0f;
    }
    for (int i = t; i < ROWS_PER_BLOCK * N_SPARSE; i += THREADS) {
        const int m = i / N_SPARSE, s = i % N_SPARSE;
        s_idx[i] = sparse[(size_t)(row0 + m) * N_SPARSE + s] + N_DENSE + s * V_DEPTH;
    }
    // B staged transposed: s_B[tcol*17 + d] = v[d*312 + tcol], zero-padded.
    // Loop order: d outer, tcol inner -> coalesced global reads, stride-17 LDS writes.
    #pragma unroll
    for (int i = t; i < NT_PAD * 16; i += THREADS) {
        const int d = i / NT_PAD, tcol = i % NT_PAD;
        s_B[tcol * BSTRIDE + d] =
            (d < N_DENSE && tcol < TK) ? v[(size_t)d * TK + tcol] : 0.0f;
    }
    __syncthreads();

    // ---------------- Phase 0b: linear term (16 threads; no barrier needed
    //                 before phase 1 — phase 1 only reads phase-0a data) ------
    if (t < ROWS_PER_BLOCK) {
        float lin = w0[0];
        #pragma unroll
        for (int d = 0; d < N_DENSE; ++d)
            lin += s_densA[t * 16 + d] * w[d];
        #pragma unroll
        for (int s = 0; s < N_SPARSE; ++s)
            lin += w[s_idx[t * N_SPARSE + s]];
        s_lin[t] = lin;
    }

    // ---------------- Phase 1: WMMA dense_part = A(16x16) x B(16x320) -------
    // 16x16x4 f32 WMMA, 4 chained k-steps; 20 waves x 1 N-tile. All operands
    // from LDS -> unconditional ds_load_b64, EXEC all-ones throughout.
    {
        const int wave  = t >> 5;          // 0..19 == ntile
        const int lane  = t & 31;
        const int hi    = lane >> 4;       // 0: lanes 0-15, 1: lanes 16-31
        const int ln    = lane & 15;
        const int tcol  = wave * 16 + ln;  // 0..319
        v8f c = {};
        #pragma unroll
        for (int ks = 0; ks < 4; ++ks) {
            const int d0 = ks * 4 + hi * 2;         // per-lane K pair
            v2f a, b;
            a[0] = s_densA[ln * 16 + d0];
            a[1] = s_densA[ln * 16 + d0 + 1];
            b[0] = s_B[tcol * BSTRIDE + d0];
            b[1] = s_B[tcol * BSTRIDE + d0 + 1];
            c = __builtin_amdgcn_wmma_f32_16x16x4_f32(
                    /*neg_a=*/false, a, /*neg_b=*/false, b,
                    /*c_mod=*/(short)0, c, /*reuse_a=*/false, /*reuse_b=*/false);
        }
        #pragma unroll
        for (int r = 0; r < 8; ++r) {
            const int m = r + 8 * hi;               // D layout: VGPR r -> M=r / M=r+8
            s_dp[m * NT_PAD + wave * 16 + ln] = c[r];
        }
    }
    __syncthreads();   // s_dp + s_lin ready; s_B dead -> s_red may reuse it

    // ---------------- Phase 2: gather-accumulate + reduction, 2 rows/iter ----
    const int pair = t / NT_PAD;          // 0 or 1 (wave-aligned split)
    const int tt   = t - pair * NT_PAD;   // 0..319 within pair group
    float* const red  = s_red  + pair * (2 * NT_PAD);
    float* const part = s_part + pair * 48;

    for (int it = 0; it < ROWS_PER_BLOCK / 2; ++it) {
        const int m = it * 2 + pair;
        if (tt < TK) {
            float acc = s_dp[m * NT_PAD + tt];
            #pragma unroll
            for (int s = 0; s < N_SPARSE; ++s) {
                const int sid = s_idx[m * N_SPARSE + s];
                acc += v[(size_t)sid * TK + tt];    // coalesced 1248B burst per index
            }
            red[tt]          = acc;
            red[NT_PAD + tt] = acc * acc;
        }
        __syncthreads();

        if (tt < K_DIM) {                           // S_k -> S_k^2
            float sk = 0.0f;
            #pragma unroll
            for (int f = 0; f < FIELD; ++f) sk += red[f * K_DIM + tt];
            part[tt] = sk * sk;
        } else if (tt >= 32 && tt < 32 + FIELD) {   // per-field sum of squares
            const int f = tt - 32;
            float q = 0.0f;
            #pragma unroll
            for (int k = 0; k < K_DIM; ++k) q += red[NT_PAD + f * K_DIM + k];
            part[K_DIM + f] = q;
        }
        __syncthreads();

        if (tt == 0) {
            float ss = 0.0f, qq = 0.0f;
            #pragma unroll
            for (int k = 0; k < K_DIM; ++k) ss += part[k];
            #pragma unroll
            for (int f = 0; f < FIELD; ++f) qq += part[K_DIM + f];
            out[row0 + m] = s_lin[m] + 0.5f * (ss - qq);
        }
        __syncthreads();  // protect red[] before next iteration
    }
}

extern "C" void kernel_launch(void* const* d_in, const int* in_sizes, int n_in,
                              void* d_out, int out_size, void* d_ws, size_t ws_size,
                              hipStream_t stream) {
    (void)in_sizes; (void)n_in; (void)d_ws; (void)ws_size; (void)out_size;
    const float* dense  = (const float*)d_in[0];
    const int*   sparse = (const int*)d_in[1];
    const float* w0     = (const float*)d_in[2];
    const float* w      = (const float*)d_in[3];
    const float* v      = (const float*)d_in[4];
    float*       out    = (float*)d_out;

    dim3 grid(BATCH / ROWS_PER_BLOCK);   // 256 blocks
    dim3 block(THREADS);                 // 640 threads = 20 wave32
    ffm_kernel<<<grid, block, 0, stream>>>(dense, sparse, w0, w, v, out);
}